// MPLayer_42494406427362
// MI455X (gfx1250) — compile-verified
//
#include <hip/hip_runtime.h>
#include <stdint.h>

// MPLayer for MI455X (gfx1250, wave32).
// Strategy: exploit linearity of the edge transform to replace the E x (64x64)
// edge GEMM with a raw-feature scatter-add (native L2 f32 atomics), then do the
// small node-side GEMMs with V_WMMA_F32_16X16X4_F32 (one wave per 16-node tile).

typedef float v2f __attribute__((ext_vector_type(2)));
typedef float v8f __attribute__((ext_vector_type(8)));

// Guaranteed-native device-scope f32 atomic add (RMW executes at L2).
__device__ __forceinline__ void atomic_add_f32_dev(float* p, float v) {
  asm volatile("global_atomic_add_f32 %0, %1, off scope:SCOPE_DEV"
               :
               : "v"((uint64_t)(uintptr_t)p), "v"(v)
               : "memory");
}

// ---------------- zero workspace ----------------
__global__ void zero_ws_kernel(float* __restrict__ ws, int n) {
  int i = blockIdx.x * blockDim.x + threadIdx.x;
  if (i < n) ws[i] = 0.0f;
}

// ---------------- scatter: S[dst] += concat(node_x[src], edge_x); deg[dst]++ --
// One wave (32 lanes) per edge; lane == feature index. edge_x / src / dst are
// streamed once -> non-temporal loads so the 26MB accumulator + 12.8MB node_x
// gather table stay resident in the 192MB L2.
__global__ __launch_bounds__(256)
void scatter_kernel(const float* __restrict__ node_x,
                    const float* __restrict__ edge_x,
                    const int* __restrict__ src,
                    const int* __restrict__ dst,
                    float* __restrict__ S,         // [N][64]
                    unsigned int* __restrict__ deg, // [N]
                    int E) {
  int e = blockIdx.x * 8 + (threadIdx.x >> 5);
  if (e >= E) return;
  int lane = threadIdx.x & 31;
  int s = __builtin_nontemporal_load(&src[e]);
  int d = __builtin_nontemporal_load(&dst[e]);
  float xv = node_x[(size_t)s * 32 + lane];                          // L2 hit
  float ev = __builtin_nontemporal_load(&edge_x[(size_t)e * 32 + lane]);
  atomic_add_f32_dev(&S[(size_t)d * 64 + lane], xv);
  atomic_add_f32_dev(&S[(size_t)d * 64 + 32 + lane], ev);
  if (lane == 0) {
    atomicAdd(&deg[d], 1u);  // native global_atomic_add_u32
  }
}

// ---------------- node pass: WMMA GEMMs ----------------
// Per wave: 16-node tile.
//   GEMM1: agg(16x64) = S_tile(16x64) @ W_pre(64x64) + deg*b_pre
//   z = deg>0 ? relu(agg) : z_init     (staged to LDS concat tile [16x96])
//   GEMM2: h(16x32) = relu(concat_tile(16x96) @ W_upd(96x32) + b_upd)
// A-matrix f32 16x4 layout: lanes 0-15 (row=lane) hold K=k0,k0+1; lanes 16-31
// hold K=k0+2,k0+3. B mirrors (rows striped across lanes per VGPR).
// C/D: VGPR v -> row v+8*half, col lane&15.
__global__ __launch_bounds__(256)
void node_kernel(const float* __restrict__ node_x,  // [N][32]
                 const float* __restrict__ z_init,  // [N][64]
                 const float* __restrict__ W_pre,   // [64][64]
                 const float* __restrict__ b_pre,   // [64]
                 const float* __restrict__ W_upd,   // [96][32]
                 const float* __restrict__ b_upd,   // [32]
                 const float* __restrict__ S,       // [N][64]
                 const unsigned int* __restrict__ deg, // [N]
                 float* __restrict__ out,           // [N][32]
                 int N) {
  __shared__ float lds[8 * 16 * 96];  // per-wave 16x96 concat tile, 48KB total
  const int wave = threadIdx.x >> 5;
  const int lane = threadIdx.x & 31;
  const int half = lane >> 4;
  const int l15 = lane & 15;
  const int base = (blockIdx.x * 8 + wave) * 16;
  if (base >= N) return;  // wave-uniform: EXEC stays all-1s for WMMA
  float* tile = &lds[wave * 16 * 96];

  // stage node_x rows into concat-tile cols [0,32)
  for (int idx = lane; idx < 16 * 32; idx += 32) {
    int r = idx >> 5, c = idx & 31;
    tile[r * 96 + c] = node_x[(size_t)(base + r) * 32 + c];
  }

  // hoist this lane's 8 row-degrees (rows v+8*half) out of the ct loop
  unsigned int dgu[8];
  float dgf[8];
#pragma unroll
  for (int v = 0; v < 8; ++v) {
    dgu[v] = deg[base + v + half * 8];
    dgf[v] = (float)dgu[v];
  }

  // ---- GEMM1 + branchless epilogue: z into concat-tile cols [32,96) ----
  for (int ct = 0; ct < 4; ++ct) {
    v8f acc = {};
#pragma unroll
    for (int t = 0; t < 16; ++t) {
      const int k = t * 4 + half * 2;
      v2f a = *(const v2f*)&S[(size_t)(base + l15) * 64 + k];
      v2f b;
      b.x = W_pre[(k + 0) * 64 + ct * 16 + l15];
      b.y = W_pre[(k + 1) * 64 + ct * 16 + l15];
      acc = __builtin_amdgcn_wmma_f32_16x16x4_f32(
          /*neg_a=*/false, a, /*neg_b=*/false, b,
          /*c_mod=*/(short)0, acc, /*reuse_a=*/false, /*reuse_b=*/false);
    }
    const float bp = b_pre[ct * 16 + l15];
#pragma unroll
    for (int v = 0; v < 8; ++v) {
      const int r = v + half * 8;
      const float zi = z_init[(size_t)(base + r) * 64 + ct * 16 + l15];
      const float av = acc[v] + dgf[v] * bp;
      const float zr = av > 0.0f ? av : 0.0f;
      tile[r * 96 + 32 + ct * 16 + l15] = (dgu[v] > 0u) ? zr : zi;  // cndmask
    }
  }

  // ---- GEMM2: out = relu(tile @ W_upd + b_upd) ----
  for (int ct = 0; ct < 2; ++ct) {
    v8f acc = {};
#pragma unroll
    for (int t = 0; t < 24; ++t) {
      const int k = t * 4 + half * 2;
      v2f a = *(const v2f*)&tile[l15 * 96 + k];
      v2f b;
      b.x = W_upd[(k + 0) * 32 + ct * 16 + l15];
      b.y = W_upd[(k + 1) * 32 + ct * 16 + l15];
      acc = __builtin_amdgcn_wmma_f32_16x16x4_f32(
          false, a, false, b, (short)0, acc, false, false);
    }
    const float bu = b_upd[ct * 16 + l15];
#pragma unroll
    for (int v = 0; v < 8; ++v) {
      const int r = v + half * 8;
      float h = acc[v] + bu;
      h = h > 0.0f ? h : 0.0f;
      __builtin_nontemporal_store(h, &out[(size_t)(base + r) * 32 + ct * 16 + l15]);
    }
  }
}

extern "C" void kernel_launch(void* const* d_in, const int* in_sizes, int n_in,
                              void* d_out, int out_size, void* d_ws, size_t ws_size,
                              hipStream_t stream) {
  const float* node_x = (const float*)d_in[0];
  const float* edge_x = (const float*)d_in[1];
  const float* z_init = (const float*)d_in[2];
  const float* W_pre  = (const float*)d_in[3];
  const float* b_pre  = (const float*)d_in[4];
  const float* W_upd  = (const float*)d_in[5];
  const float* b_upd  = (const float*)d_in[6];
  const int*   src    = (const int*)d_in[7];
  const int*   dst    = (const int*)d_in[8];

  const int N = in_sizes[0] / 32;  // node_x is [N,32]
  const int E = in_sizes[1] / 32;  // edge_x is [E,32]

  float* S = (float*)d_ws;                          // [N][64] feature accumulator
  unsigned int* deg = (unsigned int*)(S + (size_t)N * 64);  // [N] in-degree

  const int zn = N * 64 + N;  // zero S and deg (deg zeroed as float bits = 0u)
  zero_ws_kernel<<<(zn + 255) / 256, 256, 0, stream>>>(S, zn);
  scatter_kernel<<<(E + 7) / 8, 256, 0, stream>>>(node_x, edge_x, src, dst,
                                                  S, deg, E);
  node_kernel<<<(N + 127) / 128, 256, 0, stream>>>(node_x, z_init, W_pre, b_pre,
                                                   W_upd, b_upd, S, deg,
                                                   (float*)d_out, N);
}